// VAChamferLoss_80831284511318
// MI455X (gfx1250) — compile-verified
//
#include <hip/hip_runtime.h>
#include <cstdint>

#define THREADS 256
#define XPT     2               // x-points per thread (amortizes each ds_load_b128)
#define YSEG    1024            // y-points per block segment; 1024 * 16B = 16KB LDS tile

// ---------------------------------------------------------------------------
// Kernel 1: init per-point min (as +inf bit pattern) and zero freespace output
// ---------------------------------------------------------------------------
__global__ void chamfer_init_kernel(unsigned int* __restrict__ minv,
                                    float* __restrict__ out_zero, int N) {
    int i = blockIdx.x * blockDim.x + threadIdx.x;
    if (i < N) {
        minv[i]     = 0x7f800000u;   // +inf
        out_zero[i] = 0.0f;          // freespace_loss == zeros
    }
}

// ---------------------------------------------------------------------------
// Kernel 2: tiled 1-NN L1 min.  Each block: 512 x-points (2/thread) vs one
// 1024-point y-segment staged in LDS via async global->LDS DMA, then a
// conflict-free broadcast scan (uniform ds_load_b128 address per wave).
// ---------------------------------------------------------------------------
__global__ __launch_bounds__(THREADS)
void chamfer_min_kernel(const float* __restrict__ pc1,
                        const float* __restrict__ flow,
                        const float* __restrict__ pc2,
                        unsigned int* __restrict__ minv,
                        int N, int M) {
    __shared__ float4 sy[YSEG];      // padded {x,y,z,pad} per point -> ds_load_b128

    const int tid   = threadIdx.x;
    const int xbase = blockIdx.x * (THREADS * XPT);
    const int xi0   = xbase + tid;
    const int xi1   = xbase + THREADS + tid;
    const int ybase = blockIdx.y * YSEG;
    const int cnt   = min(YSEG, M - ybase);

    // --- async copy of this segment into LDS (packed 12B -> padded 16B).
    // INST_OFFSET is added to BOTH the LDS and global addresses, so one
    // (lds_off, gaddr) pair serves all 3 components of a point.
    {
        const unsigned int lds_base =
            (unsigned int)(uintptr_t)(void*)&sy[0];   // low 32 bits of shared
                                                      // aperture = LDS offset
        for (int j = tid; j < cnt; j += THREADS) {
            unsigned int lds_off = lds_base + (unsigned int)(j * 16);
            unsigned long long gaddr =
                (unsigned long long)(uintptr_t)(pc2 + (size_t)(ybase + j) * 3);
            asm volatile(
                "global_load_async_to_lds_b32 %0, %1, off\n\t"
                "global_load_async_to_lds_b32 %0, %1, off offset:4\n\t"
                "global_load_async_to_lds_b32 %0, %1, off offset:8"
                :: "v"(lds_off), "v"(gaddr)
                : "memory");
        }
    }

    // --- load my two x points while the DMA is in flight ---
    float a0 = 0.f, a1 = 0.f, a2 = 0.f;
    float b0 = 0.f, b1 = 0.f, b2 = 0.f;
    if (xi0 < N) {
        a0 = pc1[3 * xi0 + 0] + flow[3 * xi0 + 0];
        a1 = pc1[3 * xi0 + 1] + flow[3 * xi0 + 1];
        a2 = pc1[3 * xi0 + 2] + flow[3 * xi0 + 2];
    }
    if (xi1 < N) {
        b0 = pc1[3 * xi1 + 0] + flow[3 * xi1 + 0];
        b1 = pc1[3 * xi1 + 1] + flow[3 * xi1 + 1];
        b2 = pc1[3 * xi1 + 2] + flow[3 * xi1 + 2];
    }

    // --- wait for my wave's async ops, then sync the workgroup ---
    asm volatile("s_wait_asynccnt 0" ::: "memory");
    __syncthreads();

    // --- broadcast scan of the LDS tile ---
    float ma = __builtin_inff();
    float mb = __builtin_inff();
#pragma unroll 4
    for (int j = 0; j < cnt; ++j) {
        float4 y = sy[j];
        float da = fabsf(a0 - y.x) + fabsf(a1 - y.y) + fabsf(a2 - y.z);
        float db = fabsf(b0 - y.x) + fabsf(b1 - y.y) + fabsf(b2 - y.z);
        ma = fminf(ma, da);
        mb = fminf(mb, db);
    }

    // distances are >= 0 so IEEE float bits order like unsigned ints
    if (xi0 < N) atomicMin(&minv[xi0], __float_as_uint(ma));
    if (xi1 < N) atomicMin(&minv[xi1], __float_as_uint(mb));
}

// ---------------------------------------------------------------------------
// Kernel 3: mean over per-point mins -> d_out[0]
// ---------------------------------------------------------------------------
__global__ void chamfer_finalize_kernel(const unsigned int* __restrict__ minv,
                                        float* __restrict__ out, int N, float w) {
    __shared__ float red[THREADS];
    float s = 0.0f;
    for (int i = threadIdx.x; i < N; i += THREADS)
        s += __uint_as_float(minv[i]);
    red[threadIdx.x] = s;
    __syncthreads();
    for (int off = THREADS / 2; off > 0; off >>= 1) {
        if (threadIdx.x < off) red[threadIdx.x] += red[threadIdx.x + off];
        __syncthreads();
    }
    if (threadIdx.x == 0) out[0] = w * red[0] / (float)N;
}

// ---------------------------------------------------------------------------
extern "C" void kernel_launch(void* const* d_in, const int* in_sizes, int n_in,
                              void* d_out, int out_size, void* d_ws, size_t ws_size,
                              hipStream_t stream) {
    const float* pc1  = (const float*)d_in[0];
    const float* flow = (const float*)d_in[1];
    const float* pc2  = (const float*)d_in[2];
    float* out = (float*)d_out;

    const int N = in_sizes[0] / 3;   // B*N points (B == 1)
    const int M = in_sizes[2] / 3;   // B*M points

    unsigned int* minv = (unsigned int*)d_ws;   // N uints of scratch

    chamfer_init_kernel<<<(N + THREADS - 1) / THREADS, THREADS, 0, stream>>>(
        minv, out + 1, N);

    dim3 grid((N + THREADS * XPT - 1) / (THREADS * XPT), (M + YSEG - 1) / YSEG);
    chamfer_min_kernel<<<grid, THREADS, 0, stream>>>(pc1, flow, pc2, minv, N, M);

    chamfer_finalize_kernel<<<1, THREADS, 0, stream>>>(minv, out, N, 1.0f);
}